// DiagonalSSM_62302795596501
// MI455X (gfx1250) — compile-verified
//
#include <hip/hip_runtime.h>
#include <hip/hip_bf16.h>

typedef __attribute__((ext_vector_type(2))) float v2f;
typedef __attribute__((ext_vector_type(8))) float v8f;

#define DMODEL 1024
#define NSTATE 256
#define SEQ    4096
#define BATCH  8
#define BK     32
#define SAS    36              // padded LDS row stride (floats): bank-conflict-free
#define NCHUNK 64
#define CHLEN  (SEQ / NCHUNK)  // 64

// ---------------------------------------------------------------------------
// Kernel 1: alpha = sigmoid(u @ Wa^T + ba), Bu = u @ Wb^T + bb
// Block tile: 128 (M) x 32 (N). 8 waves = 4 (M) x 2 (N). Double-buffered LDS,
// one barrier per K-chunk, global loads issued one chunk ahead.
// ---------------------------------------------------------------------------
__global__ __launch_bounds__(256) void ssm_gates_kernel(
    const float* __restrict__ u, const float* __restrict__ Wa,
    const float* __restrict__ ba, const float* __restrict__ Wb,
    const float* __restrict__ bb, float* __restrict__ alpha,
    float* __restrict__ Bu)
{
    __shared__ float sA [2][128 * SAS];
    __shared__ float sWa[2][ 32 * SAS];
    __shared__ float sWb[2][ 32 * SAS];

    const int tid  = threadIdx.x;
    const int lane = tid & 31;
    const int wave = tid >> 5;
    const int wm   = wave >> 1;
    const int wn   = wave & 1;
    const int m0   = blockIdx.x * 128;
    const int n0   = blockIdx.y * 32;

    const int ar  = tid >> 3;          // 0..31
    const int ac4 = (tid & 7) << 2;    // 0,4,..,28
    const float* gA0 = u  + (size_t)(m0 + ar)      * DMODEL + ac4;
    const float* gA1 = u  + (size_t)(m0 + ar + 32) * DMODEL + ac4;
    const float* gA2 = u  + (size_t)(m0 + ar + 64) * DMODEL + ac4;
    const float* gA3 = u  + (size_t)(m0 + ar + 96) * DMODEL + ac4;
    const float* gWa = Wa + (size_t)(n0 + ar)      * DMODEL + ac4;
    const float* gWb = Wb + (size_t)(n0 + ar)      * DMODEL + ac4;
    const int sOff = ar * SAS + ac4;

    const int mr = lane & 15;
    const int kq = (lane >> 4) << 1;

    v8f acc_a0 = {}, acc_a1 = {};
    v8f acc_b0 = {}, acc_b1 = {};

    struct St { float4 a0, a1, a2, a3, wa, wb; };
    auto ld = [&](int k0) {
        St s;
        s.a0 = *(const float4*)(gA0 + k0);
        s.a1 = *(const float4*)(gA1 + k0);
        s.a2 = *(const float4*)(gA2 + k0);
        s.a3 = *(const float4*)(gA3 + k0);
        s.wa = *(const float4*)(gWa + k0);
        s.wb = *(const float4*)(gWb + k0);
        return s;
    };
    auto stg = [&](int buf, const St& s) {
        float* qa = &sA[buf][sOff];
        *(float4*)(qa)           = s.a0;
        *(float4*)(qa + 32 * SAS) = s.a1;
        *(float4*)(qa + 64 * SAS) = s.a2;
        *(float4*)(qa + 96 * SAS) = s.a3;
        *(float4*)(&sWa[buf][sOff]) = s.wa;
        *(float4*)(&sWb[buf][sOff]) = s.wb;
    };
    auto compute = [&](int buf) {
        const float* pa0 = &sA [buf][(wm * 32 + mr) * SAS + kq];
        const float* pa1 = pa0 + 16 * SAS;
        const float* pwa = &sWa[buf][(wn * 16 + mr) * SAS + kq];
        const float* pwb = &sWb[buf][(wn * 16 + mr) * SAS + kq];
#pragma unroll
        for (int kk = 0; kk < BK; kk += 4) {
            v2f a0, a1, va, vb;
            a0.x = pa0[kk]; a0.y = pa0[kk + 1];
            a1.x = pa1[kk]; a1.y = pa1[kk + 1];
            va.x = pwa[kk]; va.y = pwa[kk + 1];
            vb.x = pwb[kk]; vb.y = pwb[kk + 1];
            acc_a0 = __builtin_amdgcn_wmma_f32_16x16x4_f32(
                false, a0, false, va, (short)0, acc_a0, false, false);
            acc_a1 = __builtin_amdgcn_wmma_f32_16x16x4_f32(
                false, a1, false, va, (short)0, acc_a1, false, false);
            acc_b0 = __builtin_amdgcn_wmma_f32_16x16x4_f32(
                false, a0, false, vb, (short)0, acc_b0, false, false);
            acc_b1 = __builtin_amdgcn_wmma_f32_16x16x4_f32(
                false, a1, false, vb, (short)0, acc_b1, false, false);
        }
    };

    const int NIT = DMODEL / BK;   // 32 (even)
    St nx;
    stg(0, ld(0));
    nx = ld(BK);
    __syncthreads();
#pragma unroll 1
    for (int i = 0; i < NIT; i += 2) {
        compute(0);
        stg(1, nx);
        if (i + 2 < NIT) nx = ld((i + 2) * BK);
        __syncthreads();
        compute(1);
        if (i + 2 < NIT) {
            stg(0, nx);
            if (i + 3 < NIT) nx = ld((i + 3) * BK);
            __syncthreads();
        }
    }

    const int ncol  = n0 + wn * 16 + (lane & 15);
    const float bca = ba[ncol];
    const float bcb = bb[ncol];
    const int mb0 = m0 + wm * 32 + ((lane >> 4) << 3);
    const int mb1 = mb0 + 16;
#pragma unroll
    for (int r = 0; r < 8; ++r) {
        float a0v = acc_a0[r] + bca;
        float a1v = acc_a1[r] + bca;
        a0v = 1.0f / (1.0f + __expf(-a0v));
        a1v = 1.0f / (1.0f + __expf(-a1v));
        size_t o0 = (size_t)(mb0 + r) * NSTATE + ncol;
        size_t o1 = (size_t)(mb1 + r) * NSTATE + ncol;
        alpha[o0] = a0v;  Bu[o0] = acc_b0[r] + bcb;
        alpha[o1] = a1v;  Bu[o1] = acc_b1[r] + bcb;
    }
}

// ---------------------------------------------------------------------------
// Kernel 2: chunked parallel scan  x_s = a_s * x_{s-1} + b_s
// ---------------------------------------------------------------------------
__global__ __launch_bounds__(256) void ssm_scan1_kernel(
    const float* __restrict__ alpha, const float* __restrict__ Bu,
    float* __restrict__ Ac, float* __restrict__ Xc)
{
    const int n = threadIdx.x;
    const int c = blockIdx.x;
    const int b = blockIdx.y;
    size_t base = ((size_t)b * SEQ + (size_t)c * CHLEN) * NSTATE + n;
    float x = 0.0f, A = 1.0f;
#pragma unroll 4
    for (int s = 0; s < CHLEN; ++s) {
        float a  = alpha[base + (size_t)s * NSTATE];
        float bu = Bu   [base + (size_t)s * NSTATE];
        x = fmaf(a, x, bu);
        A *= a;
    }
    size_t idx = ((size_t)b * NCHUNK + c) * NSTATE + n;
    Ac[idx] = A;
    Xc[idx] = x;
}

__global__ __launch_bounds__(256) void ssm_scan2_kernel(
    const float* __restrict__ Ac, const float* __restrict__ Xc,
    float* __restrict__ carry)
{
    const int n = threadIdx.x;
    const int b = blockIdx.x;
    float x = 0.0f;
    for (int c = 0; c < NCHUNK; ++c) {
        size_t idx = ((size_t)b * NCHUNK + c) * NSTATE + n;
        carry[idx] = x;
        x = fmaf(Ac[idx], x, Xc[idx]);
    }
}

__global__ __launch_bounds__(256) void ssm_scan3_kernel(
    const float* __restrict__ alpha, const float* __restrict__ Bu,
    const float* __restrict__ carry, float* __restrict__ xs)
{
    const int n = threadIdx.x;
    const int c = blockIdx.x;
    const int b = blockIdx.y;
    size_t base = ((size_t)b * SEQ + (size_t)c * CHLEN) * NSTATE + n;
    float x = carry[((size_t)b * NCHUNK + c) * NSTATE + n];
#pragma unroll 4
    for (int s = 0; s < CHLEN; ++s) {
        float a  = alpha[base + (size_t)s * NSTATE];
        float bu = Bu   [base + (size_t)s * NSTATE];
        x = fmaf(a, x, bu);
        xs[base + (size_t)s * NSTATE] = x;
    }
}

// ---------------------------------------------------------------------------
// Kernel 3: y = xs @ C^T + u @ Dm^T   (C: (D,N), Dm: (D,D), row-major)
// Block tile 128 (M) x 64 (D), wave tile 32x32, double-buffered, two K phases.
// ---------------------------------------------------------------------------
__global__ __launch_bounds__(256) void ssm_out_kernel(
    const float* __restrict__ xs, const float* __restrict__ u,
    const float* __restrict__ Cm, const float* __restrict__ Dm,
    float* __restrict__ y)
{
    __shared__ float sA[2][128 * SAS];
    __shared__ float sW[2][ 64 * SAS];

    const int tid  = threadIdx.x;
    const int lane = tid & 31;
    const int wave = tid >> 5;
    const int wm   = wave >> 1;
    const int wn   = wave & 1;
    const int m0   = blockIdx.x * 128;
    const int d0   = blockIdx.y * 64;

    const int ar  = tid >> 3;
    const int ac4 = (tid & 7) << 2;
    const int sOff = ar * SAS + ac4;

    const int mr = lane & 15;
    const int kq = (lane >> 4) << 1;

    v8f acc00 = {}, acc01 = {}, acc10 = {}, acc11 = {};

    struct St { float4 a0, a1, a2, a3, w0, w1; };
    auto stg = [&](int buf, const St& s) {
        float* qa = &sA[buf][sOff];
        *(float4*)(qa)            = s.a0;
        *(float4*)(qa + 32 * SAS) = s.a1;
        *(float4*)(qa + 64 * SAS) = s.a2;
        *(float4*)(qa + 96 * SAS) = s.a3;
        float* qw = &sW[buf][sOff];
        *(float4*)(qw)            = s.w0;
        *(float4*)(qw + 32 * SAS) = s.w1;
    };
    auto compute = [&](int buf) {
        const float* pa0 = &sA[buf][(wm * 32 + mr) * SAS + kq];
        const float* pa1 = pa0 + 16 * SAS;
        const float* pw0 = &sW[buf][(wn * 32 + mr) * SAS + kq];
        const float* pw1 = pw0 + 16 * SAS;
#pragma unroll
        for (int kk = 0; kk < BK; kk += 4) {
            v2f a0, a1, w0, w1;
            a0.x = pa0[kk]; a0.y = pa0[kk + 1];
            a1.x = pa1[kk]; a1.y = pa1[kk + 1];
            w0.x = pw0[kk]; w0.y = pw0[kk + 1];
            w1.x = pw1[kk]; w1.y = pw1[kk + 1];
            acc00 = __builtin_amdgcn_wmma_f32_16x16x4_f32(
                false, a0, false, w0, (short)0, acc00, false, false);
            acc01 = __builtin_amdgcn_wmma_f32_16x16x4_f32(
                false, a0, false, w1, (short)0, acc01, false, false);
            acc10 = __builtin_amdgcn_wmma_f32_16x16x4_f32(
                false, a1, false, w0, (short)0, acc10, false, false);
            acc11 = __builtin_amdgcn_wmma_f32_16x16x4_f32(
                false, a1, false, w1, (short)0, acc11, false, false);
        }
    };

    // One phase: strides/pointers are compile-time via macro expansion.
#define OUT_PHASE(AMAT, WMAT, KD)                                             \
    {                                                                         \
        const float* gA0 = (AMAT) + (size_t)(m0 + ar)      * (KD) + ac4;      \
        const float* gA1 = (AMAT) + (size_t)(m0 + ar + 32) * (KD) + ac4;      \
        const float* gA2 = (AMAT) + (size_t)(m0 + ar + 64) * (KD) + ac4;      \
        const float* gA3 = (AMAT) + (size_t)(m0 + ar + 96) * (KD) + ac4;      \
        const float* gW0 = (WMAT) + (size_t)(d0 + ar)      * (KD) + ac4;      \
        const float* gW1 = (WMAT) + (size_t)(d0 + ar + 32) * (KD) + ac4;      \
        auto ld = [&](int k0) {                                               \
            St s;                                                             \
            s.a0 = *(const float4*)(gA0 + k0);                                \
            s.a1 = *(const float4*)(gA1 + k0);                                \
            s.a2 = *(const float4*)(gA2 + k0);                                \
            s.a3 = *(const float4*)(gA3 + k0);                                \
            s.w0 = *(const float4*)(gW0 + k0);                                \
            s.w1 = *(const float4*)(gW1 + k0);                                \
            return s;                                                         \
        };                                                                    \
        const int NIT = (KD) / BK;                                            \
        St nx;                                                                \
        stg(0, ld(0));                                                        \
        nx = ld(BK);                                                          \
        __syncthreads();                                                      \
        _Pragma("unroll 1")                                                   \
        for (int i = 0; i < NIT; i += 2) {                                    \
            compute(0);                                                       \
            stg(1, nx);                                                       \
            if (i + 2 < NIT) nx = ld((i + 2) * BK);                           \
            __syncthreads();                                                  \
            compute(1);                                                       \
            if (i + 2 < NIT) {                                                \
                stg(0, nx);                                                   \
                if (i + 3 < NIT) nx = ld((i + 3) * BK);                       \
                __syncthreads();                                              \
            }                                                                 \
        }                                                                     \
    }

    OUT_PHASE(xs, Cm, NSTATE)   // phase 1: K = 256
    OUT_PHASE(u,  Dm, DMODEL)   // phase 2: K = 1024
#undef OUT_PHASE

    const int dc0 = d0 + wn * 32 + (lane & 15);
    const int dc1 = dc0 + 16;
    const int mb0 = m0 + wm * 32 + ((lane >> 4) << 3);
    const int mb1 = mb0 + 16;
#pragma unroll
    for (int r = 0; r < 8; ++r) {
        y[(size_t)(mb0 + r) * DMODEL + dc0] = acc00[r];
        y[(size_t)(mb0 + r) * DMODEL + dc1] = acc01[r];
        y[(size_t)(mb1 + r) * DMODEL + dc0] = acc10[r];
        y[(size_t)(mb1 + r) * DMODEL + dc1] = acc11[r];
    }
}

// ---------------------------------------------------------------------------
extern "C" void kernel_launch(void* const* d_in, const int* in_sizes, int n_in,
                              void* d_out, int out_size, void* d_ws, size_t ws_size,
                              hipStream_t stream)
{
    (void)in_sizes; (void)n_in; (void)out_size; (void)ws_size;

    const float* u  = (const float*)d_in[0];
    const float* Wa = (const float*)d_in[1];
    const float* ba = (const float*)d_in[2];
    const float* Wb = (const float*)d_in[3];
    const float* bb = (const float*)d_in[4];
    const float* Cm = (const float*)d_in[5];
    const float* Dm = (const float*)d_in[6];
    float* y = (float*)d_out;

    const size_t elemsBSN = (size_t)BATCH * SEQ * NSTATE;
    const size_t elemsBCN = (size_t)BATCH * NCHUNK * NSTATE;

    float* alpha = (float*)d_ws;
    float* Bu    = alpha + elemsBSN;
    float* xsb   = Bu + elemsBSN;
    float* Ac    = xsb + elemsBSN;
    float* Xc    = Ac + elemsBCN;
    float* carry = Xc + elemsBCN;

    const int Mrows = BATCH * SEQ;   // 32768

    {
        dim3 grid(Mrows / 128, NSTATE / 32);
        ssm_gates_kernel<<<grid, 256, 0, stream>>>(u, Wa, ba, Wb, bb, alpha, Bu);
    }
    {
        dim3 g1(NCHUNK, BATCH);
        ssm_scan1_kernel<<<g1, 256, 0, stream>>>(alpha, Bu, Ac, Xc);
        ssm_scan2_kernel<<<BATCH, 256, 0, stream>>>(Ac, Xc, carry);
        ssm_scan3_kernel<<<g1, 256, 0, stream>>>(alpha, Bu, carry, xsb);
    }
    {
        dim3 grid(Mrows / 128, DMODEL / 64);
        ssm_out_kernel<<<grid, 256, 0, stream>>>(xsb, u, Cm, Dm, y);
    }
}